// EdgeLearner_32925219291944
// MI455X (gfx1250) — compile-verified
//
#include <hip/hip_runtime.h>

// EdgeLearner for MI455X (gfx1250, wave32).
// Math: softmax over w is shift-invariant per (v,l), so
//   QK^T/16  ~  h_v^T M h_w + u . h_w   with  M = Wq^T Wk /16,  u = Wk^T bq /16
// (bk and bq.h_v terms are constant in w and cancel) -> 43 GF instead of 78 GF.
// GEMMs run on V_WMMA_F32_16X16X4_F32 (fp32 reference -> fp32 matrix path).
// Stage A uses the Tensor Data Mover (tensor_load_to_lds) with LDS padding to
// land the 64x256 hs tile directly in its bank-padded LDS layout.

typedef __attribute__((ext_vector_type(2))) float v2f;
typedef __attribute__((ext_vector_type(8))) float v8f;
typedef __attribute__((ext_vector_type(4))) unsigned int v4u;
typedef __attribute__((ext_vector_type(4))) int v4i;
typedef __attribute__((ext_vector_type(8))) int v8i;

#define B_   4
#define V_   64
#define E_   1024
#define L_   1024
#define D_   256
#define DEG_ 16

#define HSB_STRIDE 260   // 256 data + 4 pad DWORDs: transposed reads stay ~conflict-free
#define T_STRIDE   260
#define S_STRIDE   68

// ---------------------------------------------------------------------------
// Precompute Mp = row-pair-interleaved (Wq^T Wk)/16 and u = Wk^T bq /16.
//   Mp[(k>>1)*512 + j*2 + (k&1)] = M[k][j]
// so a WMMA B-fragment {M[kk][j], M[kk+1][j]} is ONE aligned b64 global load.
// ---------------------------------------------------------------------------
__global__ void precompute_MU(const float* __restrict__ Wq,
                              const float* __restrict__ Wk,
                              const float* __restrict__ bq,
                              float* __restrict__ Mp,
                              float* __restrict__ uout) {
  const int i = blockIdx.x;
  const int j = threadIdx.x;
  float acc = 0.0f;
  for (int d = 0; d < D_; ++d) acc += Wq[d * D_ + i] * Wk[d * D_ + j];
  Mp[(i >> 1) * (2 * D_) + j * 2 + (i & 1)] = acc * 0.0625f;  // 1/sqrt(256)
  if (i == 0) {
    float au = 0.0f;
    for (int d = 0; d < D_; ++d) au += bq[d] * Wk[d * D_ + j];
    uout[j] = au * 0.0625f;
  }
}

// ---------------------------------------------------------------------------
// ei output: broadcast edge_index (2, B*E) over L; out[flat]=edge_index[flat/L].
// ---------------------------------------------------------------------------
__global__ void broadcast_ei(const int* __restrict__ edge_index,
                             int* __restrict__ out_ei) {
  const long long total = 2LL * B_ * E_ * L_;
  for (long long idx = (long long)blockIdx.x * 256 + threadIdx.x; idx < total;
       idx += (long long)gridDim.x * 256) {
    out_ei[idx] = edge_index[idx / L_];
  }
}

// ---------------------------------------------------------------------------
// Main kernel: one workgroup (8 wave32) per (b, l).
// ---------------------------------------------------------------------------
__launch_bounds__(256, 1)
__global__ void edge_attention(const float* __restrict__ hs,
                               const int* __restrict__ edge_index,
                               const float* __restrict__ ew,
                               const float* __restrict__ Mp,
                               const float* __restrict__ uu,
                               const float* __restrict__ skip_p,
                               float* __restrict__ out_ew) {
  __shared__ float hsb[V_ * HSB_STRIDE];   //  65 KiB
  __shared__ float Tt [V_ * T_STRIDE];     //  65 KiB
  __shared__ float Ss [V_ * S_STRIDE];     //  17 KiB
  __shared__ float hup[V_][4];
  __shared__ float hu [V_];
  __shared__ float mrow[V_];
  __shared__ float rsum[V_];

  const int l    = blockIdx.x;
  const int b    = blockIdx.y;
  const int tid  = threadIdx.x;
  const int lane = tid & 31;
  const int wv   = tid >> 5;        // wave id 0..7
  const int hh   = (lane >> 4) & 1; // half-wave (K-pair select per ISA layout)
  const int n    = lane & 15;       // row/col within fragment

  const float* base = hs + ((long long)(b * V_) * L_ + l) * D_;

  // ---- Stage A: hs[b,:,l,:] -> LDS ---------------------------------------
#if __has_builtin(__builtin_amdgcn_tensor_load_to_lds)
  // TDM: 2D tile 256x64 f32, global row stride L*D elements, LDS padding of
  // 4 DWORDs every 256 DWORDs reproduces the HSB_STRIDE=260 layout.
  if (wv == 0) {
    const unsigned long long ga = (unsigned long long)(uintptr_t)base;
    const unsigned int lds_base = (unsigned int)(uintptr_t)(&hsb[0]);
    const v4u g0 = {
        1u,                                           // count=1, user mode
        lds_base,                                     // D#.lds_addr
        (unsigned int)ga,                             // global_addr[31:0]
        ((unsigned int)(ga >> 32) & 0x1FFFFFFu) | (2u << 30)  // [56:32] | type=2
    };
    const v8i g1 = {
        (int)((2u << 16) |          // data_size = 4B
              (1u << 20) |          // pad_enable
              (7u << 22) |          // pad_interval: 256 DWORDs
              (3u << 25)),          // pad_amount:   4 DWORDs
        (int)(((unsigned)D_) << 16),   // tensor_dim0 = 256 (bits 79:48 lo)
        (int)(((unsigned)V_) << 16),   // dim0 hi = 0 | tensor_dim1 = 64 lo
        (int)(((unsigned)D_) << 16),   // dim1 hi = 0 | tile_dim0 = 256
        (int)V_,                       // tile_dim1 = 64, tile_dim2 = 0
        (int)(L_ * D_),                // tensor_dim0_stride = 262144 (lo32)
        0, 0                           // stride hi, dim1_stride (2D: unused)
    };
    const v4i gz = {0, 0, 0, 0};
#if defined(__clang_major__) && __clang_major__ >= 23
    const v8i gz8 = {0, 0, 0, 0, 0, 0, 0, 0};
    __builtin_amdgcn_tensor_load_to_lds(g0, g1, gz, gz, gz8, 0);
#else
    __builtin_amdgcn_tensor_load_to_lds(g0, g1, gz, gz, 0);
#endif
    __builtin_amdgcn_s_wait_tensorcnt(0);
  }
#else
  for (int idx = tid; idx < V_ * (D_ / 4); idx += 256) {
    const int v  = idx >> 6;
    const int c4 = idx & 63;
    const float4 val =
        *reinterpret_cast<const float4*>(base + (long long)v * (L_ * D_) + c4 * 4);
    float* dp = &hsb[v * HSB_STRIDE + c4 * 4];
    dp[0] = val.x; dp[1] = val.y; dp[2] = val.z; dp[3] = val.w;
  }
#endif
  __syncthreads();

  // ---- hu[w] = (hsb @ u)[w], 4 partials per row --------------------------
  {
    const int v = tid >> 2, part = tid & 3;
    const float* hr = &hsb[v * HSB_STRIDE + part * 64];
    const float* up = uu + part * 64;
    float acc = 0.0f;
#pragma unroll 8
    for (int j = 0; j < 64; ++j) acc += hr[j] * up[j];
    hup[v][part] = acc;
  }

  // ---- Stage B: T = hsb @ Mp  --------------------------------------------
  // Wave wv owns both col-tiles {2wv, 2wv+1} simultaneously: A-fragments
  // reused x2, B-fragments reused x4, 8 WMMAs per 2 b64 global loads.
  {
    const int jt0 = wv * 2, jt1 = jt0 + 1;
    v8f acc[2][4] = {};
    for (int k0 = 0; k0 < D_; k0 += 4) {
      const int kk = k0 + 2 * hh;                 // even
      const int pbase = (kk >> 1) * (2 * D_);
      const v2f b0 = *reinterpret_cast<const v2f*>(Mp + pbase + (jt0 * 16 + n) * 2);
      const v2f b1 = *reinterpret_cast<const v2f*>(Mp + pbase + (jt1 * 16 + n) * 2);
#pragma unroll
      for (int rt = 0; rt < 4; ++rt) {
        const v2f a = *reinterpret_cast<const v2f*>(&hsb[(rt * 16 + n) * HSB_STRIDE + kk]);
        acc[0][rt] = __builtin_amdgcn_wmma_f32_16x16x4_f32(
            false, a, false, b0, (short)0, acc[0][rt], false, false);
        acc[1][rt] = __builtin_amdgcn_wmma_f32_16x16x4_f32(
            false, a, false, b1, (short)0, acc[1][rt], false, false);
      }
    }
#pragma unroll
    for (int q = 0; q < 2; ++q)
#pragma unroll
      for (int rt = 0; rt < 4; ++rt)
#pragma unroll
        for (int r = 0; r < 8; ++r)
          Tt[(rt * 16 + r + 8 * hh) * T_STRIDE + (jt0 + q) * 16 + n] = acc[q][rt][r];
  }
  __syncthreads();

  // ---- Stage C: S = T @ hsb^T --------------------------------------------
  // Wave's two tiles share vt -> one A-fragment feeds two WMMAs.
  {
    const int vt  = wv >> 1;
    const int wt0 = (2 * wv) & 3, wt1 = wt0 + 1;
    v8f acc0 = {}, acc1 = {};
    for (int k0 = 0; k0 < D_; k0 += 4) {
      const int kk = k0 + 2 * hh;
      const v2f a   = *reinterpret_cast<const v2f*>(&Tt [(vt  * 16 + n) * T_STRIDE   + kk]);
      const v2f bf0 = *reinterpret_cast<const v2f*>(&hsb[(wt0 * 16 + n) * HSB_STRIDE + kk]);
      const v2f bf1 = *reinterpret_cast<const v2f*>(&hsb[(wt1 * 16 + n) * HSB_STRIDE + kk]);
      acc0 = __builtin_amdgcn_wmma_f32_16x16x4_f32(
          false, a, false, bf0, (short)0, acc0, false, false);
      acc1 = __builtin_amdgcn_wmma_f32_16x16x4_f32(
          false, a, false, bf1, (short)0, acc1, false, false);
    }
#pragma unroll
    for (int r = 0; r < 8; ++r) {
      Ss[(vt * 16 + r + 8 * hh) * S_STRIDE + wt0 * 16 + n] = acc0[r];
      Ss[(vt * 16 + r + 8 * hh) * S_STRIDE + wt1 * 16 + n] = acc1[r];
    }
  }
  __syncthreads();

  if (tid < V_) hu[tid] = hup[tid][0] + hup[tid][1] + hup[tid][2] + hup[tid][3];
  __syncthreads();

  // ---- Stage D: masked softmax stats per src row -------------------------
  if (tid < V_) {
    const int v = tid;
    const int* dsts = edge_index + (B_ * E_) + b * E_ + v * DEG_;
    float xs[DEG_];
    float m = -1e30f;
#pragma unroll
    for (int i = 0; i < DEG_; ++i) {
      const int d = dsts[i];
      const float x = Ss[v * S_STRIDE + d] + hu[d];
      xs[i] = x;
      m = fmaxf(m, x);
    }
    float s = 0.0f;
#pragma unroll
    for (int i = 0; i < DEG_; ++i) s += __expf(xs[i] - m);
    mrow[v] = m;
    rsum[v] = 1.0f / s;
  }
  __syncthreads();

  // ---- Stage E: out = skip*ew + (1-skip)*att -----------------------------
  {
    const float skip  = skip_p[0];
    const float oskip = 1.0f - skip;
    for (int e = tid; e < E_; e += 256) {
      const int v = e >> 4;
      const int d = edge_index[B_ * E_ + b * E_ + e];
      const float x   = Ss[v * S_STRIDE + d] + hu[d];
      const float att = __expf(x - mrow[v]) * rsum[v];
      out_ew[(long long)(b * E_ + e) * L_ + l] = skip * ew[b * E_ + e] + oskip * att;
    }
  }
}

// ---------------------------------------------------------------------------
extern "C" void kernel_launch(void* const* d_in, const int* in_sizes, int n_in,
                              void* d_out, int out_size, void* d_ws, size_t ws_size,
                              hipStream_t stream) {
  const float* hs        = (const float*)d_in[0];
  const int*   edge_idx  = (const int*)  d_in[1];
  const float* edge_w    = (const float*)d_in[2];
  const float* Wq        = (const float*)d_in[3];
  const float* bq        = (const float*)d_in[4];
  const float* Wk        = (const float*)d_in[5];
  // d_in[6] (bk) cancels inside the softmax -> unused.
  const float* attn_skip = (const float*)d_in[7];

  float* Mws = (float*)d_ws;            // 256*256 floats (pair-interleaved)
  float* uws = Mws + D_ * D_;           // 256 floats

  int*   out_ei = (int*)d_out;                          // 2*B*E*L ints
  float* out_ew = (float*)d_out + 2LL * B_ * E_ * L_;   // B*E*L floats

  precompute_MU<<<D_, D_, 0, stream>>>(Wq, Wk, bq, Mws, uws);
  broadcast_ei<<<8192, 256, 0, stream>>>(edge_idx, out_ei);
  dim3 grid(L_, B_);
  edge_attention<<<grid, 256, 0, stream>>>(hs, edge_idx, edge_w, Mws, uws,
                                           attn_skip, out_ew);
  (void)in_sizes; (void)n_in; (void)out_size; (void)ws_size;
}